// KNNEmbeddingV3_55164559949912
// MI455X (gfx1250) — compile-verified
//
#include <hip/hip_runtime.h>
#include <hip/hip_bf16.h>

// ---------------- problem constants ----------------
#define BB   8
#define NN   2048
#define DD   16
#define KK   32
#define DM   256
#define BN   (BB * NN)          // 16384 points

typedef float v2f __attribute__((ext_vector_type(2)));
typedef float v8f __attribute__((ext_vector_type(8)));
typedef unsigned u32x4 __attribute__((ext_vector_type(4)));
typedef int i32x4 __attribute__((ext_vector_type(4)));
typedef int i32x8 __attribute__((ext_vector_type(8)));

__device__ __forceinline__ v8f wmma_f32_4(v2f a, v2f b, v8f c) {
    // D = A(16x4 f32) * B(4x16 f32) + C(16x16 f32)
    return __builtin_amdgcn_wmma_f32_16x16x4_f32(
        /*neg_a=*/false, a, /*neg_b=*/false, b,
        /*c_mod=*/(short)0, c, /*reuse_a=*/false, /*reuse_b=*/false);
}

// ---- Tensor Data Mover: 2D tile (tile0 x tile1 elems of 4B) global -> LDS ----
// D# per CDNA5 ISA: group0 {count=1, lds_addr, global_addr, type=2},
// group1 {data_size=4B, tensor_dim0/1, tile_dim0/1, tensor_dim0_stride}.
__device__ __forceinline__ void tdm_load_2d_f32(const void* gaddr, unsigned lds_off,
                                                unsigned dim0, unsigned dim1,
                                                unsigned tile0, unsigned tile1,
                                                unsigned stride0 /*elements*/) {
    unsigned long long ga = (unsigned long long)gaddr;
    u32x4 g0;
    g0[0] = 1u;                                             // count=1 (valid), user mode
    g0[1] = lds_off;                                        // lds_addr (bytes)
    g0[2] = (unsigned)(ga & 0xFFFFFFFFu);                   // global_addr[31:0]
    g0[3] = (unsigned)((ga >> 32) & 0x01FFFFFFu)            // global_addr[56:32]
          | (2u << 30);                                     // type = 2 ("image")
    i32x8 g1;
    g1[0] = (int)(2u << 16);                                // workgroup_mask=0, data_size=2 (4B)
    g1[1] = (int)((dim0 & 0xFFFFu) << 16);                  // tensor_dim0[15:0]
    g1[2] = (int)((dim0 >> 16) | ((dim1 & 0xFFFFu) << 16)); // dim0[31:16] | dim1[15:0]
    g1[3] = (int)((dim1 >> 16) | (tile0 << 16));            // dim1[31:16] | tile_dim0
    g1[4] = (int)(tile1 & 0xFFFFu);                         // tile_dim1 ; tile_dim2=0
    g1[5] = (int)stride0;                                   // tensor_dim0_stride[31:0]
    g1[6] = 0;                                              // stride0 hi | stride1 lo
    g1[7] = 0;                                              // stride1 hi
    i32x4 z4 = {0, 0, 0, 0};
#if __has_include(<hip/amd_detail/amd_gfx1250_TDM.h>)
    i32x8 z8 = {0, 0, 0, 0, 0, 0, 0, 0};
    __builtin_amdgcn_tensor_load_to_lds(g0, g1, z4, z4, z8, 0);   // therock 6-arg form
#else
    __builtin_amdgcn_tensor_load_to_lds(g0, g1, z4, z4, 0);       // ROCm 7.2 5-arg form
#endif
}

// ---------------- workspace layout (float offsets) ----------------
#define OFF_MEAN   0                         // B*32
#define OFF_INV    256                       // B*32
#define OFF_PESUM  512                       // DM
#define OFF_XCRD   768                       // BN*16  raw masked coords
#define OFF_SQ     (OFF_XCRD + BN * 16)      // BN
#define OFF_RSC    (OFF_SQ + BN)             // BN  rowsum of normalized crd part
#define OFF_RSF    (OFF_RSC + BN)            // BN  rowsum of normalized ftr part
#define OFF_S      (OFF_RSF + BN)            // BN*64  GEMM A operand
#define OFF_IDX    (OFF_S + BN * 64)         // BN*32 ints

// ============ kernel 1: per-batch channel mean / 1/(std+eps) ============
__global__ void k_stats(const float* __restrict__ x, const float* __restrict__ f,
                        float* __restrict__ mean, float* __restrict__ inv) {
    int b   = blockIdx.x;
    int c   = threadIdx.x & 31;      // channel 0..31 of xc
    int sub = threadIdx.x >> 5;      // 8 row subgroups
    int ch  = c & 15;
    bool upper = (c >= 16);
    bool mask  = f[b * DD + ch] > 0.1f;
    float s = 0.f, ss = 0.f;
    for (int n = sub; n < NN; n += 8) {
        float xv = x[(b * NN + n) * DD + ch];
        float v  = upper ? (mask ? xv : 0.f) : (mask ? 0.f : xv);
        s += v; ss += v * v;
    }
    __shared__ float S1[256], S2[256];
    S1[threadIdx.x] = s; S2[threadIdx.x] = ss;
    __syncthreads();
    if (threadIdx.x < 32) {
        float ts = 0.f, tss = 0.f;
        for (int u = 0; u < 8; ++u) { ts += S1[u * 32 + threadIdx.x]; tss += S2[u * 32 + threadIdx.x]; }
        float m   = ts / (float)NN;
        float var = (tss - (float)NN * m * m) / (float)(NN - 1);  // ddof=1
        var = fmaxf(var, 0.f);
        mean[b * 32 + threadIdx.x] = m;
        inv [b * 32 + threadIdx.x] = 1.f / (sqrtf(var) + 1e-5f);
    }
}

// ============ kernel 2: per-point masked coords, |x_crd|^2, normalized row sums ============
__global__ void k_rows(const float* __restrict__ x, const float* __restrict__ f,
                       const float* __restrict__ mean, const float* __restrict__ inv,
                       float* __restrict__ xcrd, float* __restrict__ sqo,
                       float* __restrict__ rsc, float* __restrict__ rsf) {
    int r = blockIdx.x * blockDim.x + threadIdx.x;
    if (r >= BN) return;
    int b = r >> 11;
    float s = 0.f, rc = 0.f, rf = 0.f;
    #pragma unroll
    for (int d = 0; d < DD; ++d) {
        float xv  = x[r * DD + d];
        bool mask = f[b * DD + d] > 0.1f;
        float crd = mask ? 0.f : xv;
        float ftr = mask ? xv : 0.f;
        xcrd[r * DD + d] = crd;
        s += crd * crd;
        float nc = (crd - mean[b * 32 + d]) * inv[b * 32 + d];
        rc += fminf(fmaxf(nc, -10.f), 10.f);
        float nf = (ftr - mean[b * 32 + 16 + d]) * inv[b * 32 + 16 + d];
        rf += fminf(fmaxf(nf, -10.f), 10.f);
    }
    sqo[r] = s; rsc[r] = rc; rsf[r] = rf;
}

// ============ kernel 3: pe column sums + pe_loss scalar ============
__global__ void k_pe(const float* __restrict__ pc, const float* __restrict__ pf,
                     float* __restrict__ pe_sum, float* __restrict__ out_loss) {
    int t = threadIdx.x;                 // 256 threads, t == output column
    float s = 0.f, a = 0.f;
    #pragma unroll
    for (int p = 0; p < DD; ++p) {
        float v1 = pc[p * DM + t], v2 = pf[p * DM + t];
        s += v1 + v2;
        a += fabsf(v1) + fabsf(v2);
    }
    pe_sum[t] = s;
    __shared__ float red[256];
    red[t] = a; __syncthreads();
    for (int off = 128; off > 0; off >>= 1) {
        if (t < off) red[t] += red[t + off];
        __syncthreads();
    }
    if (t == 0) *out_loss = red[0];
}

// ============ kernel 4: WMMA Gram tiles -> d2 rows in LDS -> top-32 per row ============
// grid = BN/16 blocks, 512 threads (16 waves). Wave w: Gram tiles strided, then row w selection.
__global__ void k_knn(const float* __restrict__ xcrd, const float* __restrict__ sqg,
                      int* __restrict__ idxo) {
    extern __shared__ float lds[];       // 16 rows * 2048 d2 values (128 KB)
    __shared__ float sqA[16];
    int base = blockIdx.x * 16;          // global row base (same batch: N % 16 == 0)
    int b    = base >> 11;
    int n0   = base & (NN - 1);
    const float* Xb  = xcrd + (b << 11) * DD;   // [2048][16]
    const float* sqb = sqg  + (b << 11);
    int wave = threadIdx.x >> 5, lane = threadIdx.x & 31;
    int Lm = lane & 15, Lh = lane >> 4;

    if (threadIdx.x < 16) sqA[threadIdx.x] = sqb[n0 + threadIdx.x];

    // A fragment: 16 rows (this block's n-tile) x 16 dims, split into 4 K-chunks of 4
    v2f a[4];
    #pragma unroll
    for (int kk = 0; kk < 4; ++kk) {
        const float* p = &Xb[(n0 + Lm) * DD + kk * 4 + Lh * 2];
        a[kk].x = p[0]; a[kk].y = p[1];
    }
    __syncthreads();

    // Gram: each wave computes 8 of the 128 m-tiles (uniform trip count: EXEC all-1 for WMMA)
    for (int mt = wave; mt < NN / 16; mt += 16) {
        // speculative prefetch of the next m-tile this wave will touch
        __builtin_prefetch(&Xb[((((mt + 16) & 127) * 16 + Lm) * DD)], 0, 1);
        v8f c = {};
        #pragma unroll
        for (int kk = 0; kk < 4; ++kk) {
            v2f bb;
            const float* p = &Xb[(mt * 16 + Lm) * DD + kk * 4 + Lh * 2];
            bb.x = p[0]; bb.y = p[1];
            c = wmma_f32_4(a[kk], bb, c);
        }
        int   m   = mt * 16 + Lm;
        float sqm = sqb[m];
        #pragma unroll
        for (int v = 0; v < 8; ++v) {
            int row = v + 8 * Lh;                      // C layout: M = v + 8*(lane>=16)
            lds[row * NN + m] = sqA[row] + sqm - 2.f * c[v];
        }
    }
    __syncthreads();

    // Tournament top-32 (smallest d2; ties -> lower index, matching jax top_k stability).
    // Every LDS slot in this phase is read AND written by exactly one lane (m ≡ lane mod 32;
    // the removal write is done by the owning lane), so plain LDS accesses are correct —
    // no volatile (which would force FLAT SCOPE_SYS loads). Cross-lane data moves via shfl.
    float* rowp = lds + (wave << 11);
    int rglob = base + wave;
    float best = 3.402823466e38f; int besti = 1 << 30;
    for (int m = lane; m < NN; m += 32) {
        float v = rowp[m];
        if (v < best) { best = v; besti = m; }         // ascending scan keeps earliest on ties
    }
    for (int it = 0; it < KK; ++it) {
        float gb = best; int gi = besti;
        #pragma unroll
        for (int off = 16; off > 0; off >>= 1) {
            float ov = __shfl_xor(gb, off, 32);
            int   oi = __shfl_xor(gi, off, 32);
            if (ov < gb || (ov == gb && oi < gi)) { gb = ov; gi = oi; }
        }
        if (lane == 0) idxo[rglob * KK + it] = gi;
        if ((gi & 31) == lane) {                       // owning lane: remove + local rescan
            rowp[gi] = 3.402823466e38f;
            best = 3.402823466e38f; besti = 1 << 30;
            for (int m = lane; m < NN; m += 32) {
                float v = rowp[m];
                if (v < best) { best = v; besti = m; }
            }
        }
    }
}

// ============ kernel 5: S[b,n,k] from row sums + idx gather ============
__global__ void k_S(const int* __restrict__ idxo, const float* __restrict__ rsc,
                    const float* __restrict__ rsf, float* __restrict__ S) {
    int t = blockIdx.x * blockDim.x + threadIdx.x;     // BN*K threads
    if (t >= BN * KK) return;
    int r = t >> 5, k = t & 31;
    int bbase = r & ~(NN - 1);
    int j = idxo[t];
    S[r * 64 + k]      = rsc[bbase + j] - rsc[r];
    S[r * 64 + 32 + k] = rsf[bbase + j] - rsf[r];
}

// ============ kernel 6: O = S(16384x64) @ Wc(64x256) + pe_sum ============
// Block: one ntile (16 output cols); W tiles staged in LDS via TDM; 8 waves x 8 mtiles.
// B fragments loaded once per wave from LDS and reused across all 8 mtiles (128 WMMA/wave).
__global__ void k_gemm(const float* __restrict__ S, const float* __restrict__ Wcrd,
                       const float* __restrict__ Wftr, const float* __restrict__ pe_sum,
                       float* __restrict__ out) {
    __shared__ float Wsm[2][16 * 32];    // [crd/ftr][dcol_local][k]
    int wave = threadIdx.x >> 5, lane = threadIdx.x & 31;
    int ntile = blockIdx.x & 15;
    int mbase = (blockIdx.x >> 4) * 64;  // 16 mgroups x 64 mtiles
    int Lm = lane & 15, Lh = lane >> 4;

    if (threadIdx.x < 32) {
        // two contiguous-in-memory 16x32 f32 tiles (rows ntile*16..+15 of W[DM][K])
        tdm_load_2d_f32(Wcrd + ntile * 16 * KK, (unsigned)(unsigned long long)&Wsm[0][0],
                        /*dim0=*/KK, /*dim1=*/DM, /*tile0=*/KK, /*tile1=*/16, /*stride0=*/KK);
        tdm_load_2d_f32(Wftr + ntile * 16 * KK, (unsigned)(unsigned long long)&Wsm[1][0],
                        /*dim0=*/KK, /*dim1=*/DM, /*tile0=*/KK, /*tile1=*/16, /*stride0=*/KK);
        __builtin_amdgcn_s_wait_tensorcnt((short)0);
    }
    __syncthreads();

    // B fragments from LDS: Wc[k][dcol] with k rows striped over Lh, col = Lm
    v2f bfrag[16];
    #pragma unroll
    for (int kk = 0; kk < 16; ++kk) {
        int k0 = kk * 4 + Lh * 2;
        const float* src = (k0 < 32) ? &Wsm[0][Lm * KK + k0] : &Wsm[1][Lm * KK + (k0 - 32)];
        bfrag[kk].x = src[0]; bfrag[kk].y = src[1];
    }
    float pe = pe_sum[ntile * 16 + Lm];

    for (int mi = 0; mi < 8; ++mi) {
        int mtile = mbase + wave * 8 + mi;
        int arow  = mtile * 16 + Lm;
        v8f c = {};
        #pragma unroll
        for (int kk = 0; kk < 16; ++kk) {
            int k0 = kk * 4 + Lh * 2;
            v2f a;
            a.x = S[arow * 64 + k0];
            a.y = S[arow * 64 + k0 + 1];
            c = wmma_f32_4(a, bfrag[kk], c);
        }
        #pragma unroll
        for (int v = 0; v < 8; ++v) {
            int row = mtile * 16 + v + 8 * Lh;         // C layout: M = v + 8*(lane>=16)
            out[row * DM + ntile * 16 + Lm] = c[v] + pe;
        }
    }
}

// ---------------- launcher ----------------
extern "C" void kernel_launch(void* const* d_in, const int* in_sizes, int n_in,
                              void* d_out, int out_size, void* d_ws, size_t ws_size,
                              hipStream_t stream) {
    const float* x    = (const float*)d_in[0];
    const float* feat = (const float*)d_in[1];
    const float* Wcrd = (const float*)d_in[2];
    const float* Wftr = (const float*)d_in[3];
    const float* pec  = (const float*)d_in[4];
    const float* pef  = (const float*)d_in[5];
    // d_in[6] = k (compile-time constant KK here)

    float* ws    = (float*)d_ws;
    float* mean  = ws + OFF_MEAN;
    float* inv   = ws + OFF_INV;
    float* pesum = ws + OFF_PESUM;
    float* xcrd  = ws + OFF_XCRD;
    float* sq    = ws + OFF_SQ;
    float* rsc   = ws + OFF_RSC;
    float* rsf   = ws + OFF_RSF;
    float* S     = ws + OFF_S;
    int*   idx   = (int*)(ws + OFF_IDX);
    float* out   = (float*)d_out;

    k_stats<<<BB, 256, 0, stream>>>(x, feat, mean, inv);
    k_rows <<<(BN + 255) / 256, 256, 0, stream>>>(x, feat, mean, inv, xcrd, sq, rsc, rsf);
    k_pe   <<<1, 256, 0, stream>>>(pec, pef, pesum, out + (size_t)BN * DM);
    k_knn  <<<BN / 16, 512, 16 * NN * sizeof(float), stream>>>(xcrd, sq, idx);
    k_S    <<<(BN * KK + 255) / 256, 256, 0, stream>>>(idx, rsc, rsf, S);
    k_gemm <<<(1024 / 64) * 16, 256, 0, stream>>>(S, Wcrd, Wftr, pesum, out);
}